// SimpleEnergyModel_61521111548492
// MI455X (gfx1250) — compile-verified
//
#include <hip/hip_runtime.h>

typedef __attribute__((ext_vector_type(2))) float v2f;
typedef __attribute__((ext_vector_type(8))) float v8f;

#define N_ATOMS      4096
#define COULOMB_K    (-1.0f)
#define EV_CONST     (1.602e-19f)

#define H_STRIDE         18                        // 16 + pad: b64-aligned loads, fewer conflicts
#define WAVE_LDS_FLOATS  (2 * 16 * H_STRIDE + 2 * 16 * 4)  // 2 H tiles + 2 out tiles = 704 floats
#define WAVES_PER_BLOCK  8
#define BLOCK_THREADS    256
#define GRID_BLOCKS      2048
#define NWAVES           (GRID_BLOCKS * WAVES_PER_BLOCK)   // 16384 -> 64 KB partials in d_ws

__global__ __launch_bounds__(BLOCK_THREADS)
void energy_wmma_kernel(const float* __restrict__ coords,
                        const int*   __restrict__ atom_ix,
                        const float* __restrict__ charges,
                        const float* __restrict__ emb,
                        const float* __restrict__ W1,   // [2,16]
                        const float* __restrict__ b1,   // [16]
                        const float* __restrict__ W2,   // [16,3]
                        const float* __restrict__ b2,   // [3]
                        float*       __restrict__ partial)
{
    __shared__ float lds[WAVES_PER_BLOCK * WAVE_LDS_FLOATS];

    const int tid  = threadIdx.x;
    const int lane = tid & 31;
    const int wIB  = tid >> 5;
    const int half = lane >> 4;    // 0: lanes 0-15, 1: lanes 16-31
    const int l16  = lane & 15;

    float* hbufA = &lds[wIB * WAVE_LDS_FLOATS];
    float* hbufB = hbufA + 16 * H_STRIDE;
    float* obufA = hbufB + 16 * H_STRIDE;
    float* obufB = obufA + 16 * 4;

    const int waveId = blockIdx.x * WAVES_PER_BLOCK + wIB;

    // ---- wave-invariant WMMA operands ----
    // B1 = W1 padded to 4x16 (B-layout: lanes0-15 hold K=0,1 ; lanes16-31 hold K=2,3 = zeros)
    v2f B1;
    B1.x = (half == 0) ? W1[0 * 16 + l16] : 0.0f;
    B1.y = (half == 0) ? W1[1 * 16 + l16] : 0.0f;
    const float b1v = b1[l16];                     // C-layout value depends only on N = l16

    // B2 chunks: W2 padded to 16x16 (cols >= 3 zero), K split in chunks of 4
    v2f B2[4];
#pragma unroll
    for (int k = 0; k < 4; ++k) {
        const int r0 = 4 * k + 2 * half;
        B2[k].x = (l16 < 3) ? W2[(r0 + 0) * 3 + l16] : 0.0f;
        B2[k].y = (l16 < 3) ? W2[(r0 + 1) * 3 + l16] : 0.0f;
    }
    const float b2v = (l16 < 3) ? b2[l16] : 0.0f;

    float energy = 0.0f;
    const int ngroups = (N_ATOMS * N_ATOMS) / 32;  // 524,288 groups of 32 pairs

    for (int g = waveId; g < ngroups; g += NWAVES) {
        // ---- row-i data is wave-uniform: 32 pairs never straddle a row (32 | 4096) ----
        const int   i   = g >> 7;                  // (g*32) >> 12
        const float cix = coords[3 * i + 0];
        const float ciy = coords[3 * i + 1];
        const float ciz = coords[3 * i + 2];
        const int   ai  = atom_ix[i];
        const float qi  = charges[i];
        const int   tribase = ai * (ai + 1) / 2;

        // ---- per-lane gather: pair p = g*32 + lane, all 32 lanes active ----
        const int   j  = ((g & 127) << 5) + lane;  // (g*32) & 4095 + lane
        const float dx = cix - coords[3 * j + 0];
        const float dy = ciy - coords[3 * j + 1];
        const float dz = ciz - coords[3 * j + 2];
        const float sq = dx * dx + dy * dy + dz * dz;
        const float rd = (sq > 0.0f) ? fminf(1.0f / sqrtf(sq), 10.0f) : 10.0f;
        const float e  = emb[tribase + atom_ix[j]];
        const float q  = qi * charges[j];

        // move high-half inputs into lanes 0-15 for the second A tile
        const float e_hi = __shfl_down(e, 16, 32);
        const float q_hi = __shfl_down(q, 16, 32);

        // A-layout: lanes 0-15 hold (K=0,K=1); lanes 16-31 hold (K=2,K=3) = zeros
        v2f A1a, A1b;
        A1a.x = (half == 0) ? e    : 0.0f;  A1a.y = (half == 0) ? q    : 0.0f;
        A1b.x = (half == 0) ? e_hi : 0.0f;  A1b.y = (half == 0) ? q_hi : 0.0f;

        // ---- layer 1: two independent 16x16x4 WMMAs ----
        v8f acc1a, acc1b;
#pragma unroll
        for (int r = 0; r < 8; ++r) { acc1a[r] = b1v; acc1b[r] = b1v; }
        acc1a = __builtin_amdgcn_wmma_f32_16x16x4_f32(false, A1a, false, B1,
                                                      (short)0, acc1a, false, false);
        acc1b = __builtin_amdgcn_wmma_f32_16x16x4_f32(false, A1b, false, B1,
                                                      (short)0, acc1b, false, false);

        // ReLU + store H tiles (D-layout: M = r + 8*half on VGPRs, N = l16 on lanes)
#pragma unroll
        for (int r = 0; r < 8; ++r) {
            hbufA[(r + 8 * half) * H_STRIDE + l16] = fmaxf(acc1a[r], 0.0f);
            hbufB[(r + 8 * half) * H_STRIDE + l16] = fmaxf(acc1b[r], 0.0f);
        }

        __builtin_amdgcn_wave_barrier();           // cross-lane LDS handoff

        // ---- layer 2: two interleaved chains of 4 WMMAs (K = 16) ----
        v8f acc2a, acc2b;
#pragma unroll
        for (int r = 0; r < 8; ++r) { acc2a[r] = b2v; acc2b[r] = b2v; }
#pragma unroll
        for (int k = 0; k < 4; ++k) {
            v2f A2a, A2b;                          // A-layout: M = l16, K = 4k + 2*half (+1)
            const int ko = 4 * k + 2 * half;
            A2a.x = hbufA[l16 * H_STRIDE + ko + 0];
            A2a.y = hbufA[l16 * H_STRIDE + ko + 1];
            A2b.x = hbufB[l16 * H_STRIDE + ko + 0];
            A2b.y = hbufB[l16 * H_STRIDE + ko + 1];
            acc2a = __builtin_amdgcn_wmma_f32_16x16x4_f32(false, A2a, false, B2[k],
                                                          (short)0, acc2a, false, false);
            acc2b = __builtin_amdgcn_wmma_f32_16x16x4_f32(false, A2b, false, B2[k],
                                                          (short)0, acc2b, false, false);
        }

        // ---- scatter the 3 useful output columns, transposed ----
        if (l16 < 3) {
#pragma unroll
            for (int r = 0; r < 8; ++r) {
                obufA[(r + 8 * half) * 4 + l16] = acc2a[r];
                obufB[(r + 8 * half) * 4 + l16] = acc2b[r];
            }
        }

        __builtin_amdgcn_wave_barrier();

        // ---- epilogue: all 32 lanes, lane L owns pair L of this group ----
        {
            const float* ob = (half == 0) ? obufA : obufB;
            const float c_w = ob[l16 * 4 + 0];
            const float sig = ob[l16 * 4 + 1];
            const float r_w = ob[l16 * 4 + 2];
            const float sr  = sig * rd;
            const float sr2 = sr * sr;
            const float sr6 = sr2 * sr2 * sr2;
            energy += COULOMB_K * (c_w * rd) + EV_CONST * (r_w * (sr6 * sr6 - sr6));
        }

        __builtin_amdgcn_wave_barrier();           // WAR on obuf before next iteration
    }

    // full-wave reduction
#pragma unroll
    for (int s = 16; s > 0; s >>= 1)
        energy += __shfl_down(energy, s, 32);
    if (lane == 0)
        partial[waveId] = energy;
}

__global__ __launch_bounds__(256)
void reduce_kernel(const float* __restrict__ partial,
                   const float* __restrict__ bias,
                   float*       __restrict__ out)
{
    __shared__ float sm[256];
    float s = 0.0f;
    for (int i = threadIdx.x; i < NWAVES; i += 256)
        s += partial[i];
    sm[threadIdx.x] = s;
    __syncthreads();
    for (int w = 128; w > 0; w >>= 1) {
        if (threadIdx.x < w) sm[threadIdx.x] += sm[threadIdx.x + w];
        __syncthreads();
    }
    if (threadIdx.x == 0)
        out[0] = sm[0] + bias[0];
}

extern "C" void kernel_launch(void* const* d_in, const int* in_sizes, int n_in,
                              void* d_out, int out_size, void* d_ws, size_t ws_size,
                              hipStream_t stream)
{
    (void)in_sizes; (void)n_in; (void)out_size; (void)ws_size;

    const float* coords  = (const float*)d_in[0];
    const int*   atom_ix = (const int*)  d_in[1];
    const float* charges = (const float*)d_in[2];
    const float* emb     = (const float*)d_in[3];
    const float* W1      = (const float*)d_in[4];
    const float* b1      = (const float*)d_in[5];
    const float* W2      = (const float*)d_in[6];
    const float* b2      = (const float*)d_in[7];
    const float* bias    = (const float*)d_in[8];

    float* partial = (float*)d_ws;                 // NWAVES floats = 64 KB
    float* out     = (float*)d_out;

    energy_wmma_kernel<<<GRID_BLOCKS, BLOCK_THREADS, 0, stream>>>(
        coords, atom_ix, charges, emb, W1, b1, W2, b2, partial);
    reduce_kernel<<<1, 256, 0, stream>>>(partial, bias, out);
}